// RelativePositionEncoding_87995289960492
// MI455X (gfx1250) — compile-verified
//
#include <hip/hip_runtime.h>
#include <hip/hip_bf16.h>
#include <stdint.h>

// RelativePositionEncoding: out[i,j,:] = W[clip(ri[i]-ri[j]+32, 0, 64)] + b
// Store-bandwidth bound: 1.208 GB f32 out, ~52 us floor @ 23.3 TB/s; inputs L2-resident.
// CDNA5 paths: two TDM tensor_load_to_lds DMAs (W table + residue_index -> LDS),
// s_wait_tensorcnt, wave32 tiling (1 wave == one 512B contiguous output row slice),
// nontemporal b128 streaming stores (output is 6x L2 capacity).

#define NUM_BINS    32
#define NUM_CLASSES 65
#define PAIR_DIM    128
#define MAX_N       2048

typedef float        v4f   __attribute__((ext_vector_type(4)));
typedef unsigned int u32x4 __attribute__((ext_vector_type(4)));
typedef int          i32x4 __attribute__((ext_vector_type(4)));
typedef int          i32x8 __attribute__((ext_vector_type(8)));

#if __has_builtin(__builtin_amdgcn_tensor_load_to_lds)
// Issue one TDM 2D tile load: rows x cols elements of 4B, row stride = cols,
// contiguous source at gaddr, destination LDS byte offset laddr.
__device__ __forceinline__ void tdm_load_2d_f32(uint64_t gaddr, uint32_t laddr,
                                                unsigned cols, unsigned rows) {
    u32x4 g0;
    g0.x = 1u;                                      // count=1 (valid), user mode, no gather
    g0.y = laddr;                                   // lds_addr (bytes)
    g0.z = (uint32_t)(gaddr & 0xFFFFFFFFu);         // global_addr[31:0]
    g0.w = (uint32_t)((gaddr >> 32) & 0x1FFFFFFu)   // global_addr[56:32]
         | (2u << 30);                              // D# type = 2 ("image")

    i32x8 g1;
    g1[0] = (int)(2u << 16);                        // workgroup_mask=0; data_size=2 -> 4B
    g1[1] = (int)(cols << 16);                      // tensor_dim0 = cols
    g1[2] = (int)(rows << 16);                      // tensor_dim1 = rows
    g1[3] = (int)(cols << 16);                      // tile_dim0   = cols
    g1[4] = (int)rows;                              // tile_dim1   = rows, tile_dim2 = 0
    g1[5] = (int)cols;                              // tensor_dim0_stride = cols
    g1[6] = 0;
    g1[7] = 0;
    i32x4 g2 = {0, 0, 0, 0};                        // no dim2/3, no iterate
    i32x4 g3 = {0, 0, 0, 0};
    i32x8 g4 = {0, 0, 0, 0, 0, 0, 0, 0};            // extra group (6-arg toolchain): zero

    __builtin_amdgcn_tensor_load_to_lds(g0, g1, g2, g3, g4, 0);
}
#endif

__global__ __launch_bounds__(256)
void relpos_embed_kernel(const int* __restrict__ ri,
                         const float* __restrict__ W,
                         const float* __restrict__ bias,
                         float* __restrict__ out,
                         int N) {
    __shared__ __align__(16) float Tw[NUM_CLASSES * PAIR_DIM]; // 33,280 B: raw W table
    __shared__ __align__(16) int   Ri[MAX_N];                  // residue_index staged

    const int tid = threadIdx.x;
    const int i   = blockIdx.x;

#if __has_builtin(__builtin_amdgcn_tensor_load_to_lds)
    // Wave 0 DMAs both tables into LDS; other waves go straight to the barrier.
    if (tid < 32) {
        tdm_load_2d_f32((uint64_t)(uintptr_t)W,
                        (uint32_t)(uintptr_t)(void*)Tw, PAIR_DIM, NUM_CLASSES);
        tdm_load_2d_f32((uint64_t)(uintptr_t)ri,
                        (uint32_t)(uintptr_t)(void*)Ri, (unsigned)N, 1u);
        __builtin_amdgcn_s_wait_tensorcnt(0);       // TENSORcnt == 0 before barrier signal
    }
#else
    for (int idx = tid; idx < NUM_CLASSES * PAIR_DIM; idx += 256) Tw[idx] = W[idx];
    for (int idx = tid; idx < N; idx += 256)                      Ri[idx] = ri[idx];
#endif

    __syncthreads();

    // wave w of 8 handles j in [w*jPer, ...); lane owns channels [4*lane, 4*lane+4).
    const int wv   = tid >> 5;
    const int ln   = tid & 31;
    const int jPer = (N + 7) >> 3;
    const int j0   = wv * jPer;
    int       j1   = j0 + jPer; if (j1 > N) j1 = N;

    const v4f bv   = *(const v4f*)(bias + 4 * ln);  // bias lives in registers
    const int ri_i = Ri[i];
    float* outRow  = out + (size_t)i * (size_t)N * PAIR_DIM + (size_t)(4 * ln);

    #pragma unroll 4
    for (int j = j0; j < j1; ++j) {
        int d = ri_i - Ri[j] + NUM_BINS;            // uniform across the wave
        d = d < 0 ? 0 : d;
        d = d > 2 * NUM_BINS ? 2 * NUM_BINS : d;    // -> v_med3-style clamp
        const v4f w4 = *(const v4f*)(Tw + d * PAIR_DIM + 4 * ln);  // ds_load_b128
        const v4f v  = w4 + bv;
        __builtin_nontemporal_store(v, (v4f*)(outRow + (size_t)j * PAIR_DIM));
    }
}

extern "C" void kernel_launch(void* const* d_in, const int* in_sizes, int n_in,
                              void* d_out, int out_size, void* d_ws, size_t ws_size,
                              hipStream_t stream) {
    const int*   ri  = (const int*)d_in[0];    // [1, N] int32
    const float* W   = (const float*)d_in[1];  // [65, 128] f32
    const float* b   = (const float*)d_in[2];  // [128] f32
    float*       out = (float*)d_out;          // [1, N, N, 128] f32
    const int N = in_sizes[0];                 // B == 1 in the reference

    relpos_embed_kernel<<<dim3(N), dim3(256), 0, stream>>>(ri, W, b, out, N);
}